// GCNNSingleKernel_71305047048579
// MI455X (gfx1250) — compile-verified
//
#include <hip/hip_runtime.h>
#include <hip/hip_bf16.h>

typedef __attribute__((ext_vector_type(2))) float v2f;
typedef __attribute__((ext_vector_type(8))) float v8f;

#define EPSN 1e-5f

// ---------------------------------------------------------------------------
// sq[b*N+n] = sum_d X[b][d][n]^2
__global__ void sq_kernel(const float* __restrict__ X, float* __restrict__ sq,
                          int D, int N) {
    int i = blockIdx.x * 256 + threadIdx.x;   // over B*N
    int b = i / N, n = i % N;
    float acc = 0.f;
    for (int d = 0; d < D; ++d) {
        float x = X[(b * D + d) * N + n];
        acc += x * x;
    }
    sq[i] = acc;
}

// ---------------------------------------------------------------------------
// Fused: adjacency tile (WMMA gram) -> exp -> deg partial + msg = E @ A (WMMA)
// One wave owns 16 output columns (m). Block = 8 waves = 128 columns.
// n-loop split over gridDim.y partial buffers for occupancy + determinism.
template <int DPAD, int FPAD>
__global__ __launch_bounds__(256)
void fused_adj_msg(const float* __restrict__ X, int Dreal,
                   const float* __restrict__ sq,
                   const float* __restrict__ E, int Freal,
                   const float* __restrict__ sigma,   // null => inv_s2 = 1
                   float* __restrict__ msg_part, float* __restrict__ deg_part,
                   int N) {
    constexpr int XP = DPAD + 1;   // padded LDS strides (bank-conflict free)
    constexpr int EP = FPAD + 1;
    __shared__ float Xs[16 * XP];
    __shared__ float Es[16 * EP];
    __shared__ float SQs[16];

    const int tid  = threadIdx.x;
    const int lane = tid & 31;
    const int wave = tid >> 5;
    const int b      = blockIdx.z;
    const int Bn     = gridDim.z;
    const int split  = blockIdx.y;
    const int NSPLIT = gridDim.y;
    const int m_base = blockIdx.x * 128 + wave * 16;
    const int mloc   = lane & 15;
    const int mcol   = m_base + mloc;
    const int kb     = (lane >> 4) * 2;   // K sub-pair per half-wave
    const int hi     = lane >> 4;

    float inv_s2 = 1.0f;
    if (sigma) { float s = sigma[0]; inv_s2 = 1.0f / (s * s); }

    // Preload gram B-operand: B(k, m) = X[d][mcol]; constant over n-loop.
    v2f bg[DPAD / 4];
#pragma unroll
    for (int ks = 0; ks < DPAD / 4; ++ks) {
        int d0 = ks * 4 + kb;
        bg[ks].x = (d0     < Dreal) ? X[(b * Dreal + d0    ) * N + mcol] : 0.f;
        bg[ks].y = (d0 + 1 < Dreal) ? X[(b * Dreal + d0 + 1) * N + mcol] : 0.f;
    }
    const float sqm = sq[b * N + mcol];

    v8f zeroV = {};
    v8f acc[FPAD / 16];
#pragma unroll
    for (int fb = 0; fb < FPAD / 16; ++fb) acc[fb] = zeroV;
    float dacc = 0.f;

    const int n_begin = split * (N / NSPLIT);
    const int n_end   = n_begin + (N / NSPLIT);
    for (int nb = n_begin; nb < n_end; nb += 16) {
        // --- cooperative tile loads into LDS -------------------------------
        if (tid < 16 * DPAD) {
            int nl = tid / DPAD, d = tid % DPAD;
            Xs[nl * XP + d] = (d < Dreal) ? X[(b * Dreal + d) * N + nb + nl] : 0.f;
        }
        if (tid < 16) SQs[tid] = sq[b * N + nb + tid];
        for (int idx = tid; idx < 16 * FPAD; idx += 256) {
            int f = idx >> 4, nl = idx & 15;
            Es[nl * EP + f] = (f < Freal) ? E[(b * Freal + f) * N + nb + nl] : 0.f;
        }
        __syncthreads();

        // --- gram tile: G[n][m] = sum_d X[d][n]*X[d][m] via f32 WMMA -------
        v8f G = zeroV;
#pragma unroll
        for (int ks = 0; ks < DPAD / 4; ++ks) {
            v2f a;
            a.x = Xs[mloc * XP + ks * 4 + kb];       // A(n_loc, k) = X[k][n]
            a.y = Xs[mloc * XP + ks * 4 + kb + 1];
            G = __builtin_amdgcn_wmma_f32_16x16x4_f32(false, a, false, bg[ks],
                                                      (short)0, G, false, false);
        }

        // --- adjacency = exp(-max(sq[n]+sq[m]-2G,0)*inv_s2); deg partial ---
#pragma unroll
        for (int v = 0; v < 8; ++v) {
            float sqn = SQs[v + 8 * hi];
            float d2  = fmaxf(sqn + sqm - 2.0f * G[v], 0.f);
            float aij = __expf(-d2 * inv_s2);
            G[v] = aij;
            dacc += aij;
        }

        // --- msg += E(F x 16n) @ A(16n x 16m): 4 K-steps of 4 -------------
        // Re-layout C-layout adjacency regs into B-operand via half-wave swap.
#pragma unroll
        for (int si = 0; si < 4; ++si) {
            const int s  = si * 4;
            const int a0 = s & 4;
            float gx0 = G[a0], gy0 = G[a0 + 1], gx1 = G[a0 + 2], gy1 = G[a0 + 3];
            v2f bb;
            if (s < 8) {
                float rx = __shfl_xor(gx1, 16, 32);
                float ry = __shfl_xor(gy1, 16, 32);
                bb.x = (lane < 16) ? gx0 : rx;
                bb.y = (lane < 16) ? gy0 : ry;
            } else {
                float rx = __shfl_xor(gx0, 16, 32);
                float ry = __shfl_xor(gy0, 16, 32);
                bb.x = (lane < 16) ? rx : gx1;
                bb.y = (lane < 16) ? ry : gy1;
            }
#pragma unroll
            for (int fb = 0; fb < FPAD / 16; ++fb) {
                v2f ae;   // A(f_loc, k) = E[f][nb+s+k]
                ae.x = Es[(s + kb    ) * EP + fb * 16 + mloc];
                ae.y = Es[(s + kb + 1) * EP + fb * 16 + mloc];
                acc[fb] = __builtin_amdgcn_wmma_f32_16x16x4_f32(
                    false, ae, false, bb, (short)0, acc[fb], false, false);
            }
        }
        __syncthreads();
    }

    // --- write partials ----------------------------------------------------
#pragma unroll
    for (int fb = 0; fb < FPAD / 16; ++fb)
#pragma unroll
        for (int v = 0; v < 8; ++v) {
            int f = fb * 16 + v + 8 * hi;
            msg_part[((size_t)(split * Bn + b) * FPAD + f) * N + mcol] = acc[fb][v];
        }
    float dtot = dacc + __shfl_xor(dacc, 16, 32);
    if (lane < 16) deg_part[(size_t)(split * Bn + b) * N + mcol] = dtot;
}

// ---------------------------------------------------------------------------
__global__ void reduce4_kernel(const float* __restrict__ part,
                               float* __restrict__ out, int chunk) {
    int i = blockIdx.x * 256 + threadIdx.x;
    if (i < chunk)
        out[i] = part[i] + part[chunk + i] + part[2 * chunk + i] + part[3 * chunk + i];
}

// ---------------------------------------------------------------------------
// out[b][o][n] = relu(deg[n]*<w[o][:F],emb> + <w[o][F:],msg> + b[o])
//             + <wres[o],emb> + bres[o]
__global__ void gopconv_kernel(const float* __restrict__ emb, int Fin,
                               const float* __restrict__ msg, int FmsgRows,
                               const float* __restrict__ deg,
                               const float* __restrict__ w,
                               const float* __restrict__ bias,
                               const float* __restrict__ wres,
                               const float* __restrict__ bres,
                               float* __restrict__ out, int N) {
    int n = blockIdx.x * 256 + threadIdx.x;
    int o = blockIdx.y, Fout = gridDim.y, b = blockIdx.z;
    float s1 = 0.f, s2 = 0.f, s3 = 0.f;
    for (int f = 0; f < Fin; ++f) {
        float e = emb[(b * Fin + f) * N + n];
        s1 += w[o * 2 * Fin + f] * e;
        s2 += w[o * 2 * Fin + Fin + f] * msg[((size_t)b * FmsgRows + f) * N + n];
        s3 += wres[o * Fin + f] * e;
    }
    float conv = deg[b * N + n] * s1 + s2 + bias[o];
    out[((size_t)b * Fout + o) * N + n] = fmaxf(conv, 0.f) + s3 + bres[o];
}

// ---------------------------------------------------------------------------
// spatialnorm over node dim per (b,f): one block per (b,f)
__global__ void snorm_kernel(const float* __restrict__ in,
                             float* __restrict__ out, int N) {
    int bf = blockIdx.x, t = threadIdx.x;
    float s1 = 0.f, s2 = 0.f;
    for (int n = t; n < N; n += 256) {
        float x = in[(size_t)bf * N + n];
        s1 += x; s2 += x * x;
    }
    __shared__ float sa[256], sb[256];
    sa[t] = s1; sb[t] = s2;
    __syncthreads();
    for (int o = 128; o > 0; o >>= 1) {
        if (t < o) { sa[t] += sa[t + o]; sb[t] += sb[t + o]; }
        __syncthreads();
    }
    float m   = sa[0] / (float)N;
    float var = sb[0] / (float)N - m * m;
    float inv = rsqrtf(var + EPSN);
    for (int n = t; n < N; n += 256)
        out[(size_t)bf * N + n] = (in[(size_t)bf * N + n] - m) * inv;
}

// ---------------------------------------------------------------------------
// proj[b][d][n] = sum_f W[d][f] * emb[b][f][n];  one block per (b,d), DK=16,F=64
__global__ void proj_kernel(const float* __restrict__ W,
                            const float* __restrict__ emb,
                            float* __restrict__ proj, int N) {
    int bd = blockIdx.x;
    int b = bd >> 4, d = bd & 15, t = threadIdx.x;
    __shared__ float wr[64];
    if (t < 64) wr[t] = W[d * 64 + t];
    __syncthreads();
    for (int n = t; n < N; n += 256) {
        float acc = 0.f;
#pragma unroll 8
        for (int f = 0; f < 64; ++f) acc += wr[f] * emb[((size_t)b * 64 + f) * N + n];
        proj[((size_t)b * 16 + d) * N + n] = acc;
    }
}

// ---------------------------------------------------------------------------
__global__ void pool_kernel(const float* __restrict__ emb,
                            float* __restrict__ pooled, int N) {
    int bf = blockIdx.x, t = threadIdx.x;
    float acc = 0.f;
    for (int n = t; n < N; n += 256) acc += emb[(size_t)bf * N + n];
    __shared__ float s[256];
    s[t] = acc;
    __syncthreads();
    for (int o = 128; o > 0; o >>= 1) {
        if (t < o) s[t] += s[t + o];
        __syncthreads();
    }
    if (t == 0) pooled[bf] = s[0] / (float)N;
}

// ---------------------------------------------------------------------------
// InstanceNorm over F=64 per sample + fc + sigmoid.  One block (64 thr) per b.
__global__ void final_kernel(const float* __restrict__ pooled,
                             const float* __restrict__ fw,
                             const float* __restrict__ fb,
                             float* __restrict__ out) {
    __shared__ float s[64];
    int b = blockIdx.x, t = threadIdx.x;
    float x = pooled[b * 64 + t];
    s[t] = x; __syncthreads();
    for (int o = 32; o > 0; o >>= 1) { if (t < o) s[t] += s[t + o]; __syncthreads(); }
    float m = s[0] / 64.f;
    __syncthreads();
    float d = x - m;
    s[t] = d * d; __syncthreads();
    for (int o = 32; o > 0; o >>= 1) { if (t < o) s[t] += s[t + o]; __syncthreads(); }
    float var = s[0] / 64.f;
    __syncthreads();
    float nrm = d * rsqrtf(var + EPSN);
    s[t] = nrm * fw[t]; __syncthreads();
    for (int o = 32; o > 0; o >>= 1) { if (t < o) s[t] += s[t + o]; __syncthreads(); }
    if (t == 0) out[b] = 1.f / (1.f + __expf(-(s[0] + fb[0])));
}

// ---------------------------------------------------------------------------
extern "C" void kernel_launch(void* const* d_in, const int* in_sizes, int n_in,
                              void* d_out, int out_size, void* d_ws, size_t ws_size,
                              hipStream_t stream) {
    (void)in_sizes; (void)n_in; (void)out_size; (void)ws_size;
    const int B = 4, N = 2048, F0 = 6, F = 64, DK = 16, L = 2, NSPLIT = 4;

    const float* emb_in   = (const float*)d_in[0];
    const float* sigma    = (const float*)d_in[1];
    const float* fst_w    = (const float*)d_in[2];
    const float* fst_b    = (const float*)d_in[3];
    const float* fst_wres = (const float*)d_in[4];
    const float* fst_bres = (const float*)d_in[5];
    const float* adj_proj = (const float*)d_in[6];
    const float* w        = (const float*)d_in[7];
    const float* bconv    = (const float*)d_in[8];
    const float* wres     = (const float*)d_in[9];
    const float* bres     = (const float*)d_in[10];
    const float* fcl_w    = (const float*)d_in[11];
    const float* fcl_b    = (const float*)d_in[12];

    float* ws = (float*)d_ws;
    size_t off = 0;
    float* embA  = ws + off; off += (size_t)B * F * N;
    float* embB  = ws + off; off += (size_t)B * F * N;
    float* normE = ws + off; off += (size_t)B * F * N;
    float* msg   = ws + off; off += (size_t)B * F * N;
    float* proj  = ws + off; off += (size_t)B * DK * N;
    float* sqb   = ws + off; off += (size_t)B * N;
    float* deg   = ws + off; off += (size_t)B * N;
    float* pooled= ws + off; off += (size_t)B * F;
    float* msg_p = ws + off; off += (size_t)NSPLIT * B * F * N;
    float* deg_p = ws + off; off += (size_t)NSPLIT * B * N;

    dim3 blk(256);

    // ---- layer 0: gaussian adjacency from raw inputs -----------------------
    sq_kernel<<<dim3((B * N) / 256), blk, 0, stream>>>(emb_in, sqb, F0, N);
    fused_adj_msg<8, 16><<<dim3(N / 128, NSPLIT, B), blk, 0, stream>>>(
        emb_in, F0, sqb, emb_in, F0, sigma, msg_p, deg_p, N);
    reduce4_kernel<<<dim3((B * 16 * N) / 256), blk, 0, stream>>>(msg_p, msg, B * 16 * N);
    reduce4_kernel<<<dim3((B * N) / 256), blk, 0, stream>>>(deg_p, deg, B * N);
    gopconv_kernel<<<dim3(N / 256, F, B), blk, 0, stream>>>(
        emb_in, F0, msg, 16, deg, fst_w, fst_b, fst_wres, fst_bres, embA, N);

    // ---- learned-adjacency layers -----------------------------------------
    float* cur = embA; float* nxt = embB;
    for (int i = 0; i < L; ++i) {
        proj_kernel<<<dim3(B * DK), blk, 0, stream>>>(
            adj_proj + (size_t)i * DK * F, cur, proj, N);
        sq_kernel<<<dim3((B * N) / 256), blk, 0, stream>>>(proj, sqb, DK, N);
        snorm_kernel<<<dim3(B * F), blk, 0, stream>>>(cur, normE, N);
        fused_adj_msg<16, 64><<<dim3(N / 128, NSPLIT, B), blk, 0, stream>>>(
            proj, DK, sqb, normE, F, nullptr, msg_p, deg_p, N);
        reduce4_kernel<<<dim3((B * F * N) / 256), blk, 0, stream>>>(msg_p, msg, B * F * N);
        reduce4_kernel<<<dim3((B * N) / 256), blk, 0, stream>>>(deg_p, deg, B * N);
        gopconv_kernel<<<dim3(N / 256, F, B), blk, 0, stream>>>(
            normE, F, msg, F, deg, w + (size_t)i * F * 2 * F, bconv + (size_t)i * F,
            wres + (size_t)i * F * F, bres + (size_t)i * F, nxt, N);
        float* t = cur; cur = nxt; nxt = t;
    }

    // ---- readout -----------------------------------------------------------
    pool_kernel<<<dim3(B * F), blk, 0, stream>>>(cur, pooled, N);
    final_kernel<<<dim3(B), dim3(64), 0, stream>>>(pooled, fcl_w, fcl_b, (float*)d_out);
}